// BiTFMP_72782515798969
// MI455X (gfx1250) — compile-verified
//
#include <hip/hip_runtime.h>
#include <hip/hip_bf16.h>
#include <math.h>

// Problem constants (match reference setup_inputs)
#define BB 8
#define CC 128
#define NN 256
#define KK 96
#define AA 512
#define NA 768      // NN + AA
#define ME 960      // 2*KK + NA (joint edge count)
#define EPS_ 1e-8f

typedef __attribute__((ext_vector_type(2))) float v2f;
typedef __attribute__((ext_vector_type(8))) float v8f;

// Global-address-space pointer types: force global_load/global_store (not flat)
typedef __attribute__((address_space(1))) float       GF;
typedef __attribute__((address_space(1))) const float GCF;
typedef __attribute__((address_space(1))) const v2f   GCV2;

// ---------------------------------------------------------------------------
// Batched strided WMMA fp32 GEMM: D[b,i,j] = sum_k A[b,i,k]*s[b,k]*B[b,k,j]
// One wave per 16x64 output strip (4 accumulators share each A fragment).
// All strides and K are compile-time template parameters, and base pointers
// are immutable: every load is global_load from a fixed base + immediate
// offset, so loads clause-group ahead of the V_WMMA_F32_16X16X4_F32 chain
// with no per-iteration address arithmetic.
// ---------------------------------------------------------------------------
template <bool AC, bool BC, bool SCALED, int KDIM,
          long SAB, long SAR, long SAC,
          long SBB, long SBR, long SBC,
          long SSB, long SDB, int LDD>
__global__ __launch_bounds__(32)
void wmma_gemm_f32(const float* __restrict__ Am, const float* __restrict__ Bm,
                   const float* __restrict__ scaleK, float* __restrict__ Dm)
{
    const int lane = threadIdx.x;      // 0..31
    const int half = lane >> 4;        // 0 | 1  (K-pair select)
    const int lrow = lane & 15;        // row (A) / col (B,D) within tile
    const int b    = blockIdx.z;
    const long i0  = (long)blockIdx.y * 16;
    const long j0  = (long)blockIdx.x * 64;

    GCF* pA  = (GCF*)Am + (long)b * SAB + (i0 + lrow) * SAR + (long)(2 * half) * SAC;
    GCF* pB0 = (GCF*)Bm + (long)b * SBB + (j0 +  0 + lrow) * SBC + (long)(2 * half) * SBR;
    GCF* pB1 = pB0 + 16 * SBC;
    GCF* pB2 = pB0 + 32 * SBC;
    GCF* pB3 = pB0 + 48 * SBC;
    GCF* pS  = SCALED ? ((GCF*)scaleK + (long)b * SSB + 2 * half) : (GCF*)0;

    v8f acc0 = {}, acc1 = {}, acc2 = {}, acc3 = {};

#pragma unroll 4
    for (int k = 0; k < KDIM; k += 4) {
        const long ka = (long)k * SAC;
        const long kb = (long)k * SBR;
        float a0, a1;
        if (AC) { v2f t = *(GCV2*)(pA + ka); a0 = t[0]; a1 = t[1]; }
        else    { a0 = pA[ka]; a1 = pA[ka + SAC]; }
        if (SCALED) { a0 *= pS[k]; a1 *= pS[k + 1]; }
        v2f av = {a0, a1};

        float b00, b01, b10, b11, b20, b21, b30, b31;
        if (BC) {
            v2f t0 = *(GCV2*)(pB0 + kb); b00 = t0[0]; b01 = t0[1];
            v2f t1 = *(GCV2*)(pB1 + kb); b10 = t1[0]; b11 = t1[1];
            v2f t2 = *(GCV2*)(pB2 + kb); b20 = t2[0]; b21 = t2[1];
            v2f t3 = *(GCV2*)(pB3 + kb); b30 = t3[0]; b31 = t3[1];
        } else {
            b00 = pB0[kb]; b01 = pB0[kb + SBR];
            b10 = pB1[kb]; b11 = pB1[kb + SBR];
            b20 = pB2[kb]; b21 = pB2[kb + SBR];
            b30 = pB3[kb]; b31 = pB3[kb + SBR];
        }
        v2f bv0 = {b00, b01}, bv1 = {b10, b11}, bv2 = {b20, b21}, bv3 = {b30, b31};
        acc0 = __builtin_amdgcn_wmma_f32_16x16x4_f32(false, av, false, bv0, (short)0, acc0, false, false);
        acc1 = __builtin_amdgcn_wmma_f32_16x16x4_f32(false, av, false, bv1, (short)0, acc1, false, false);
        acc2 = __builtin_amdgcn_wmma_f32_16x16x4_f32(false, av, false, bv2, (short)0, acc2, false, false);
        acc3 = __builtin_amdgcn_wmma_f32_16x16x4_f32(false, av, false, bv3, (short)0, acc3, false, false);
    }

    // C/D layout: VGPR v -> row = v + 8*half, col = lrow
    GF* Db = (GF*)Dm + (long)b * SDB + (i0 + 8 * half) * (long)LDD + j0 + lrow;
#pragma unroll
    for (int v = 0; v < 8; ++v) {
        Db[(long)v * LDD +  0] = acc0[v];
        Db[(long)v * LDD + 16] = acc1[v];
        Db[(long)v * LDD + 32] = acc2[v];
        Db[(long)v * LDD + 48] = acc3[v];
    }
}

// ---------------------------------------------------------------------------
// Elementwise / reduction helper kernels
// ---------------------------------------------------------------------------
__global__ void fill_zero(float* __restrict__ p, long n) {
    long i  = (long)blockIdx.x * blockDim.x + threadIdx.x;
    long st = (long)gridDim.x * blockDim.x;
    for (; i < n; i += st) ((GF*)p)[i] = 0.0f;
}

// L2-normalize features along C (per b,n)
__global__ void norm_feat(const float* __restrict__ f, float* __restrict__ Xn) {
    int t = blockIdx.x * blockDim.x + threadIdx.x;
    if (t >= BB * NN) return;
    int b = t / NN, n = t % NN;
    GCF* base = (GCF*)f + (long)b * CC * NN + n;
    float s = 0.0f;
    for (int c = 0; c < CC; ++c) { float v = base[(long)c * NN]; s += v * v; }
    float inv = 1.0f / fmaxf(sqrtf(s), 1e-12f);
    GF* o = (GF*)Xn + (long)b * CC * NN + n;
    for (int c = 0; c < CC; ++c) o[(long)c * NN] = base[(long)c * NN] * inv;
}

// L2-normalize memory rows along C (per a)
__global__ void norm_mem(const float* __restrict__ m, float* __restrict__ Mn) {
    int a = blockIdx.x * blockDim.x + threadIdx.x;
    if (a >= AA) return;
    GCF* base = (GCF*)m + (long)a * CC;
    float s = 0.0f;
    for (int c = 0; c < CC; ++c) { float v = base[c]; s += v * v; }
    float inv = 1.0f / fmaxf(sqrtf(s), 1e-12f);
    GF* o = (GF*)Mn + (long)a * CC;
    for (int c = 0; c < CC; ++c) o[c] = base[c] * inv;
}

// top-4 (descending, lowest index wins ties) over a strided vector
__global__ void topk4_kernel(const float* __restrict__ S, int* __restrict__ idx,
                             int totalOuter, int outerPerBatch,
                             long batchStride, long outerStride, long elemStride,
                             int len)
{
    int t = blockIdx.x * blockDim.x + threadIdx.x;
    if (t >= totalOuter) return;
    int b = t / outerPerBatch, o = t % outerPerBatch;
    GCF* base = (GCF*)S + (long)b * batchStride + (long)o * outerStride;
    float bv[4] = {-3.4e38f, -3.4e38f, -3.4e38f, -3.4e38f};
    int   bi[4] = {0, 0, 0, 0};
    for (int e = 0; e < len; ++e) {
        float v = base[(long)e * elemStride];
        if (v > bv[3]) {
            int p = 3;
            while (p > 0 && v > bv[p - 1]) { bv[p] = bv[p - 1]; bi[p] = bi[p - 1]; --p; }
            bv[p] = v; bi[p] = e;
        }
    }
    idx[(long)t * 4 + 0] = bi[0]; idx[(long)t * 4 + 1] = bi[1];
    idx[(long)t * 4 + 2] = bi[2]; idx[(long)t * 4 + 3] = bi[3];
}

// H[:, 0:N, 0:K] = adj
__global__ void set_adj(const float* __restrict__ adj, float* __restrict__ H) {
    int t = blockIdx.x * blockDim.x + threadIdx.x;
    if (t >= BB * NN * KK) return;
    int b = t / (NN * KK), r = t % (NN * KK), n = r / KK, k = r % KK;
    ((GF*)H)[(long)b * NA * ME + (long)n * ME + k] = ((GCF*)adj)[t];
}

// H[:, N:, K:2K] = mem_adj (batch-broadcast)
__global__ void set_mem(const float* __restrict__ ma, float* __restrict__ H) {
    int t = blockIdx.x * blockDim.x + threadIdx.x;
    if (t >= BB * AA * KK) return;
    int b = t / (AA * KK), r = t % (AA * KK), a = r / KK, k = r % KK;
    ((GF*)H)[(long)b * NA * ME + (long)(NN + a) * ME + (KK + k)] = ((GCF*)ma)[(long)a * KK + k];
}

// forward cross edges: col 2K+n holds {n} U {N+knn[b,n,0..3]}
__global__ void set_cross_fwd(const int* __restrict__ knn, float* __restrict__ H) {
    int t = blockIdx.x * blockDim.x + threadIdx.x;
    if (t >= BB * NN) return;
    int b = t / NN, n = t % NN;
    GF* base = (GF*)H + (long)b * NA * ME;
    long col = 2 * KK + n;
    base[(long)n * ME + col] = 1.0f;
#pragma unroll
    for (int j = 0; j < 4; ++j) {
        int m = knn[(long)t * 4 + j];
        base[(long)(NN + m) * ME + col] = 1.0f;
    }
}

// reverse cross edges: col 2K+N+a holds {N+a} U {rev[b,a,0..3]}
__global__ void set_cross_rev(const int* __restrict__ rev, float* __restrict__ H) {
    int t = blockIdx.x * blockDim.x + threadIdx.x;
    if (t >= BB * AA) return;
    int b = t / AA, a = t % AA;
    GF* base = (GF*)H + (long)b * NA * ME;
    long col = 2 * KK + NN + a;
    base[(long)(NN + a) * ME + col] = 1.0f;
#pragma unroll
    for (int j = 0; j < 4; ++j) {
        int n = rev[(long)t * 4 + j];
        base[(long)n * ME + col] = 1.0f;
    }
}

// edge degree inverse: deinv[b,m] = 1/(sum_r H[b,r,m] + eps)
__global__ void colsum_deinv(const float* __restrict__ H, float* __restrict__ deinv) {
    int t = blockIdx.x * blockDim.x + threadIdx.x;
    if (t >= BB * ME) return;
    int b = t / ME, m = t % ME;
    GCF* base = (GCF*)H + (long)b * NA * ME + m;
    float s = 0.0f;
    for (int r = 0; r < NA; ++r) s += base[(long)r * ME];
    ((GF*)deinv)[t] = 1.0f / (s + EPS_);
}

// dis[b,i] = rsqrt( (sum_j W[b,i,j]) + 1 + eps )   (the +1 is the identity)
__global__ void rowsum_dis(const float* __restrict__ W, float* __restrict__ dis) {
    int t = blockIdx.x * blockDim.x + threadIdx.x;
    if (t >= BB * NA) return;
    int b = t / NA, i = t % NA;
    GCF* base = (GCF*)W + (long)b * NA * NA + (long)i * NA;
    float s = 1.0f;
    for (int j = 0; j < NA; ++j) s += base[j];
    ((GF*)dis)[t] = rsqrtf(s + EPS_);
}

// Amat = D^-1/2 (W + I) D^-1/2, in place on W
__global__ void amat_scale(float* __restrict__ W, const float* __restrict__ dis) {
    long i  = (long)blockIdx.x * blockDim.x + threadIdx.x;
    long st = (long)gridDim.x * blockDim.x;
    const long tot = (long)BB * NA * NA;
    GF* w = (GF*)W;
    GCF* d = (GCF*)dis;
    for (; i < tot; i += st) {
        long b = i / ((long)NA * NA);
        long r = i % ((long)NA * NA);
        int  ri = (int)(r / NA), ci = (int)(r % NA);
        float val = w[i] + (ri == ci ? 1.0f : 0.0f);
        w[i] = d[b * NA + ri] * val * d[b * NA + ci];
    }
}

// Y0[b,i,c] = joint_X^T : features for i<N, total_feats for i>=N
__global__ void build_y0(const float* __restrict__ f, const float* __restrict__ mem,
                         float* __restrict__ Y0) {
    long t = (long)blockIdx.x * blockDim.x + threadIdx.x;
    if (t >= (long)BB * NA * CC) return;
    long b = t / ((long)NA * CC);
    long r = t % ((long)NA * CC);
    int  i = (int)(r / CC), c = (int)(r % CC);
    ((GF*)Y0)[t] = (i < NN) ? ((GCF*)f)[b * CC * NN + (long)c * NN + i]
                            : ((GCF*)mem)[(long)(i - NN) * CC + c];
}

// X_hat[b,c,n] = 0.9*Y0 + 0.09*Y1 + 0.009*Y2 + 0.001*Y3   (rows n < N)
__global__ void write_xhat(const float* __restrict__ Y0, const float* __restrict__ Y1,
                           const float* __restrict__ Y2, const float* __restrict__ Y3,
                           float* __restrict__ out) {
    long t = (long)blockIdx.x * blockDim.x + threadIdx.x;
    if (t >= (long)BB * CC * NN) return;
    long b = t / ((long)CC * NN);
    long r = t % ((long)CC * NN);
    int  c = (int)(r / NN), n = (int)(r % NN);
    long y = b * NA * CC + (long)n * CC + c;
    ((GF*)out)[t] = 0.9f * ((GCF*)Y0)[y] + 0.09f * ((GCF*)Y1)[y]
                  + 0.009f * ((GCF*)Y2)[y] + 0.001f * ((GCF*)Y3)[y];
}

// per-edge / per-node degree scalings for the edge readout
__global__ void edge_deg(const float* __restrict__ adj, float* __restrict__ inv_e) {
    int t = blockIdx.x * blockDim.x + threadIdx.x;
    if (t >= BB * KK) return;
    int b = t / KK, k = t % KK;
    GCF* base = (GCF*)adj + (long)b * NN * KK + k;
    float s = 0.0f;
    for (int n = 0; n < NN; ++n) s += base[(long)n * KK];
    ((GF*)inv_e)[t] = 1.0f / fmaxf(s, EPS_);
}
__global__ void node_deg(const float* __restrict__ adj, float* __restrict__ inv_n) {
    int t = blockIdx.x * blockDim.x + threadIdx.x;
    if (t >= BB * NN) return;
    int b = t / NN, n = t % NN;
    GCF* base = (GCF*)adj + (long)b * NN * KK + (long)n * KK;
    float s = 0.0f;
    for (int k = 0; k < KK; ++k) s += base[k];
    ((GF*)inv_n)[t] = rsqrtf(fmaxf(s, EPS_));
}

// X_E[b,k,c] = inv_e[b,k] * sum_n adj[b,n,k]*inv_n[b,n]*X_hat[b,c,n]
__global__ void edge_readout(const float* __restrict__ adj, const float* __restrict__ inv_n,
                             const float* __restrict__ inv_e, const float* __restrict__ xhat,
                             float* __restrict__ xE) {
    int t = blockIdx.x * blockDim.x + threadIdx.x;
    if (t >= BB * KK * CC) return;
    int b = t / (KK * CC), r = t % (KK * CC), k = r / CC, c = r % CC;
    GCF* ab = (GCF*)adj + (long)b * NN * KK + k;
    GCF* nb = (GCF*)inv_n + (long)b * NN;
    GCF* xb = (GCF*)xhat + (long)b * CC * NN + (long)c * NN;
    float acc = 0.0f;
    for (int n = 0; n < NN; ++n)
        acc += ab[(long)n * KK] * nb[n] * xb[n];
    ((GF*)xE)[t] = ((GCF*)inv_e)[(long)b * KK + k] * acc;
}

// ---------------------------------------------------------------------------
// Orchestration
// ---------------------------------------------------------------------------
static inline int cdiv(long a, int b) { return (int)((a + b - 1) / b); }

extern "C" void kernel_launch(void* const* d_in, const int* in_sizes, int n_in,
                              void* d_out, int out_size, void* d_ws, size_t ws_size,
                              hipStream_t stream) {
    const float* features    = (const float*)d_in[0]; // (B,C,N)
    const float* adj         = (const float*)d_in[1]; // (B,N,K)
    const float* total_feats = (const float*)d_in[2]; // (A,C)
    const float* mem_adj     = (const float*)d_in[3]; // (A,K)

    float* out_xhat = (float*)d_out;                        // (B,C,N)
    float* out_xE   = out_xhat + (long)BB * CC * NN;        // (B,K,C)

    // Workspace layout (floats)
    float* ws = (float*)d_ws;
    long o = 0;
    float* Xn    = ws + o; o += (long)BB * CC * NN;   // normalized features
    float* Mn    = ws + o; o += (long)AA * CC;        // normalized memory
    float* sim   = ws + o; o += (long)BB * NN * AA;   // cosine similarities
    float* H     = ws + o; o += (long)BB * NA * ME;   // joint incidence
    float* deinv = ws + o; o += (long)BB * ME;        // 1/(edge degree)
    float* W     = ws + o; o += (long)BB * NA * NA;   // clique expansion -> Amat
    float* dis   = ws + o; o += (long)BB * NA;        // D^-1/2
    float* Y0    = ws + o; o += (long)BB * NA * CC;   // X^T and Krylov chain
    float* Y1    = ws + o; o += (long)BB * NA * CC;
    float* Y2    = ws + o; o += (long)BB * NA * CC;
    float* Y3    = ws + o; o += (long)BB * NA * CC;
    float* inv_e = ws + o; o += (long)BB * KK;
    float* inv_n = ws + o; o += (long)BB * NN;
    int* knn = (int*)(ws + o);                        // (B,N,4)
    int* rev = knn + (long)BB * NN * 4;               // (B,A,4)
    long need_bytes = o * sizeof(float) + ((long)BB * NN * 4 + (long)BB * AA * 4) * sizeof(int);
    if ((long)ws_size < need_bytes) return;           // insufficient scratch

    const dim3 blk(256);

    // 1) normalize
    norm_feat<<<cdiv(BB * NN, 256), blk, 0, stream>>>(features, Xn);
    norm_mem<<<cdiv(AA, 256), blk, 0, stream>>>(total_feats, Mn);

    // 2) sim[b,n,a] = sum_c Xn[b,c,n] * Mn[a,c]   (WMMA, M=256 N=512 K=128)
    wmma_gemm_f32<false, true, false, CC,
                  (long)CC * NN, 1L, (long)NN,
                  0L, 1L, (long)CC,
                  0L, (long)NN * AA, AA>
        <<<dim3(AA / 64, NN / 16, BB), dim3(32), 0, stream>>>(Xn, Mn, nullptr, sim);

    // 3) top-4 both directions
    topk4_kernel<<<cdiv(BB * NN, 256), blk, 0, stream>>>(
        sim, knn, BB * NN, NN, (long)NN * AA, (long)AA, 1L, AA);
    topk4_kernel<<<cdiv(BB * AA, 256), blk, 0, stream>>>(
        sim, rev, BB * AA, AA, (long)NN * AA, 1L, (long)AA, NN);

    // 4) build joint incidence H (B,768,960)
    fill_zero<<<2048, blk, 0, stream>>>(H, (long)BB * NA * ME);
    set_adj<<<cdiv((long)BB * NN * KK, 256), blk, 0, stream>>>(adj, H);
    set_mem<<<cdiv((long)BB * AA * KK, 256), blk, 0, stream>>>(mem_adj, H);
    set_cross_fwd<<<cdiv(BB * NN, 256), blk, 0, stream>>>(knn, H);
    set_cross_rev<<<cdiv(BB * AA, 256), blk, 0, stream>>>(rev, H);

    // 5) edge-degree inverse, then W = (H diag(deinv)) H^T  (WMMA, K=960)
    colsum_deinv<<<cdiv(BB * ME, 256), blk, 0, stream>>>(H, deinv);
    wmma_gemm_f32<true, true, true, ME,
                  (long)NA * ME, (long)ME, 1L,
                  (long)NA * ME, 1L, (long)ME,
                  (long)ME, (long)NA * NA, NA>
        <<<dim3(NA / 64, NA / 16, BB), dim3(32), 0, stream>>>(H, H, deinv, W);

    // 6) symmetric normalization -> Amat (in place on W)
    rowsum_dis<<<cdiv(BB * NA, 256), blk, 0, stream>>>(W, dis);
    amat_scale<<<2048, blk, 0, stream>>>(W, dis);

    // 7) Krylov chain: Y0 = X^T, Y_{l+1} = Amat Y_l  (WMMA, M=768 N=128 K=768)
    build_y0<<<cdiv((long)BB * NA * CC, 256), blk, 0, stream>>>(features, total_feats, Y0);
    const float* Yin[3]  = {Y0, Y1, Y2};
    float*       Yout[3] = {Y1, Y2, Y3};
    for (int l = 0; l < 3; ++l) {
        wmma_gemm_f32<true, false, false, NA,
                      (long)NA * NA, (long)NA, 1L,
                      (long)NA * CC, (long)CC, 1L,
                      0L, (long)NA * CC, CC>
            <<<dim3(CC / 64, NA / 16, BB), dim3(32), 0, stream>>>(W, Yin[l], nullptr, Yout[l]);
    }

    // 8) X_hat = polynomial combine (rows n < N), transpose back to (B,C,N)
    write_xhat<<<cdiv((long)BB * CC * NN, 256), blk, 0, stream>>>(Y0, Y1, Y2, Y3, out_xhat);

    // 9) edge readout X_E
    edge_deg<<<cdiv(BB * KK, 256), blk, 0, stream>>>(adj, inv_e);
    node_deg<<<cdiv(BB * NN, 256), blk, 0, stream>>>(adj, inv_n);
    edge_readout<<<cdiv(BB * KK * CC, 256), blk, 0, stream>>>(adj, inv_n, inv_e,
                                                              out_xhat, out_xE);
}